// MQATransformer_48052094108187
// MI455X (gfx1250) — compile-verified
//
#include <hip/hip_runtime.h>
#include <math.h>

typedef float v2f __attribute__((ext_vector_type(2)));
typedef float v8f __attribute__((ext_vector_type(8)));

#define LNUM 4
#define NH   16
#define DH   64
#define HD   1024
#define FF   4096
#define BB   2
#define SS   2048
#define TOK  (BB * SS)          // 4096 tokens
#define QKVN (HD + 2 * DH)      // 1152

__device__ __forceinline__ v8f wmma_f32(v2f a, v2f b, v8f c) {
  // D = A(16x4,f32) * B(4x16,f32) + C(16x16,f32)
  return __builtin_amdgcn_wmma_f32_16x16x4_f32(
      /*neg_a=*/false, a, /*neg_b=*/false, b,
      /*c_mod=*/(short)0, c, /*reuse_a=*/false, /*reuse_b=*/false);
}

// ---------------------------------------------------------------- embedding
__global__ void __launch_bounds__(256)
embed_kernel(const int* __restrict__ ids, const int* __restrict__ pos,
             const float* __restrict__ wte, const float* __restrict__ wpe,
             float* __restrict__ h) {
  int t = blockIdx.x;
  int i = threadIdx.x;                 // 256 threads * float4 = 1024 floats
  int id = ids[t];
  int p  = pos[t];
  float4 a = ((const float4*)(wte + (size_t)id * HD))[i];
  float4 b = ((const float4*)(wpe + (size_t)p  * HD))[i];
  float4 r;
  r.x = a.x + b.x; r.y = a.y + b.y; r.z = a.z + b.z; r.w = a.w + b.w;
  ((float4*)(h + (size_t)t * HD))[i] = r;
}

// ---------------------------------------------------------------- layernorm
__global__ void __launch_bounds__(256)
ln_kernel(const float* __restrict__ xin, const float* __restrict__ w,
          const float* __restrict__ bb, float* __restrict__ y) {
  __shared__ float r1[8], r2[8];
  int t = blockIdx.x, i = threadIdx.x;
  float4 v = ((const float4*)(xin + (size_t)t * HD))[i];
  float s = v.x + v.y + v.z + v.w;
#pragma unroll
  for (int o = 16; o > 0; o >>= 1) s += __shfl_xor(s, o, 32);
  if ((i & 31) == 0) r1[i >> 5] = s;
  __syncthreads();
  float mu = 0.f;
#pragma unroll
  for (int k = 0; k < 8; k++) mu += r1[k];
  mu *= (1.0f / HD);
  float dx = v.x - mu, dy = v.y - mu, dz = v.z - mu, dw = v.w - mu;
  float s2 = dx * dx + dy * dy + dz * dz + dw * dw;
#pragma unroll
  for (int o = 16; o > 0; o >>= 1) s2 += __shfl_xor(s2, o, 32);
  if ((i & 31) == 0) r2[i >> 5] = s2;
  __syncthreads();
  float var = 0.f;
#pragma unroll
  for (int k = 0; k < 8; k++) var += r2[k];
  var *= (1.0f / HD);
  float rs = rsqrtf(var + 1e-5f);
  float4 wv = ((const float4*)w)[i];
  float4 bv = ((const float4*)bb)[i];
  float4 o4;
  o4.x = dx * rs * wv.x + bv.x;
  o4.y = dy * rs * wv.y + bv.y;
  o4.z = dz * rs * wv.z + bv.z;
  o4.w = dw * rs * wv.w + bv.w;
  ((float4*)(y + (size_t)t * HD))[i] = o4;
}

// ---------------------------------------------------------------- GEMM
// C[M,N] = A[M,K] @ W[N,K]^T + bias  (+residual | +gelu)
// block = 256 thr = 8 waves; block tile 128(M) x 64(N); each wave 16x64.
// Manually software-pipelined (unroll 2, prefetch distance 2 k-steps):
// two distinct operand buffer sets so the allocator cannot collapse the
// rotation; each buffer is refilled right after its WMMAs consume it.
// K must be a multiple of 8 (true for 1024 and 4096).
__global__ void __launch_bounds__(256)
gemm_kernel(const float* __restrict__ A, const float* __restrict__ W,
            const float* __restrict__ bias, const float* __restrict__ res,
            float* __restrict__ C, int N, int K, int epi) {
  int tN   = blockIdx.x * 64;
  int tM   = blockIdx.y * 128;
  int wave = threadIdx.x >> 5;
  int lane = threadIdx.x & 31;
  int m16  = lane & 15;
  int g    = lane >> 4;
  int row0 = tM + wave * 16;

  const float* arow = A + (size_t)(row0 + m16) * K + 2 * g;
  const float* w0   = W + (size_t)(tN + m16) * K + 2 * g;
  const float* w1   = w0 + (size_t)16 * K;
  const float* w2   = w1 + (size_t)16 * K;
  const float* w3   = w2 + (size_t)16 * K;

  v8f acc0 = {0,0,0,0,0,0,0,0}, acc1 = {0,0,0,0,0,0,0,0};
  v8f acc2 = {0,0,0,0,0,0,0,0}, acc3 = {0,0,0,0,0,0,0,0};

  // Buffer set 0 (k), buffer set 1 (k+4).
  v2f xa  = *(const v2f*)(arow);
  v2f xb0 = *(const v2f*)(w0);
  v2f xb1 = *(const v2f*)(w1);
  v2f xb2 = *(const v2f*)(w2);
  v2f xb3 = *(const v2f*)(w3);
  v2f ya  = *(const v2f*)(arow + 4);
  v2f yb0 = *(const v2f*)(w0 + 4);
  v2f yb1 = *(const v2f*)(w1 + 4);
  v2f yb2 = *(const v2f*)(w2 + 4);
  v2f yb3 = *(const v2f*)(w3 + 4);

#pragma unroll 1
  for (int k0 = 8; k0 < K; k0 += 8) {
    // stage A: consume buffer0, refill it from k0
    acc0 = wmma_f32(xa, xb0, acc0);
    acc1 = wmma_f32(xa, xb1, acc1);
    acc2 = wmma_f32(xa, xb2, acc2);
    acc3 = wmma_f32(xa, xb3, acc3);
    xa  = *(const v2f*)(arow + k0);
    xb0 = *(const v2f*)(w0 + k0);
    xb1 = *(const v2f*)(w1 + k0);
    xb2 = *(const v2f*)(w2 + k0);
    xb3 = *(const v2f*)(w3 + k0);
    // stage B: consume buffer1, refill it from k0+4
    acc0 = wmma_f32(ya, yb0, acc0);
    acc1 = wmma_f32(ya, yb1, acc1);
    acc2 = wmma_f32(ya, yb2, acc2);
    acc3 = wmma_f32(ya, yb3, acc3);
    ya  = *(const v2f*)(arow + k0 + 4);
    yb0 = *(const v2f*)(w0 + k0 + 4);
    yb1 = *(const v2f*)(w1 + k0 + 4);
    yb2 = *(const v2f*)(w2 + k0 + 4);
    yb3 = *(const v2f*)(w3 + k0 + 4);
  }
  // epilogue: drain both buffers
  acc0 = wmma_f32(xa, xb0, acc0);
  acc1 = wmma_f32(xa, xb1, acc1);
  acc2 = wmma_f32(xa, xb2, acc2);
  acc3 = wmma_f32(xa, xb3, acc3);
  acc0 = wmma_f32(ya, yb0, acc0);
  acc1 = wmma_f32(ya, yb1, acc1);
  acc2 = wmma_f32(ya, yb2, acc2);
  acc3 = wmma_f32(ya, yb3, acc3);

  v8f accs[4] = {acc0, acc1, acc2, acc3};
#pragma unroll
  for (int jn = 0; jn < 4; jn++) {
    int col = tN + jn * 16 + m16;
    float bsv = bias[col];
#pragma unroll
    for (int r = 0; r < 8; r++) {
      int row = row0 + r + 8 * g;          // C layout: m = r + 8*g, n = lane&15
      float v = accs[jn][r] + bsv;
      if (epi == 1) v += res[(size_t)row * N + col];
      if (epi == 2) v = 0.5f * v * (1.0f + erff(v * 0.70710678118654752f));
      C[(size_t)row * N + col] = v;
    }
  }
}

// ---------------------------------------------------------------- attention
// MQA flash attention. Grid: (S/16 query blocks, B). Block = 512 thr = 16
// waves, one head per wave. Scores computed transposed: S^T = K(16t x D) x Q^T.
// Out accumulated transposed: out^T = V^T x P, P moved C-layout -> B-layout
// with shfl_xor(16) (cross half-wave).
__global__ void __launch_bounds__(512)
attn_kernel(const float* __restrict__ qkv, float* __restrict__ out,
            float scale, float unscale) {
  int b    = blockIdx.y;
  int s0   = blockIdx.x * 16;
  int head = threadIdx.x >> 5;
  int lane = threadIdx.x & 31;
  int n    = lane & 15;          // query column (B/C) and A row index
  int g    = lane >> 4;

  const float* base = qkv + (size_t)b * SS * QKVN;

  // Preload Q^T in B-layout for all 16 K-slices of D=64.
  const float* qrow = base + (size_t)(s0 + n) * QKVN + head * DH + 2 * g;
  v2f qb[16];
#pragma unroll
  for (int ks = 0; ks < 16; ks++) qb[ks] = *(const v2f*)(qrow + ks * 4);

  float run_m = -3.0e38f, run_s = 0.0f;
  v8f acc[4] = {{0,0,0,0,0,0,0,0},{0,0,0,0,0,0,0,0},
                {0,0,0,0,0,0,0,0},{0,0,0,0,0,0,0,0}};

  int ntiles = blockIdx.x + 1;   // causal: key tiles t0 <= s0
  for (int kt = 0; kt < ntiles; kt++) {
    int t0 = kt * 16;
    // ---- scores^T tile: A = K rows (key t0+n), B = preloaded Q^T.
    // Burst all 16 A loads first, then run the WMMA chain.
    const float* krow = base + (size_t)(t0 + n) * QKVN + HD + 2 * g;
    v2f ka[16];
#pragma unroll
    for (int ks = 0; ks < 16; ks++) ka[ks] = *(const v2f*)(krow + ks * 4);
    v8f st = {0,0,0,0,0,0,0,0};
#pragma unroll
    for (int ks = 0; ks < 16; ks++) st = wmma_f32(ka[ks], qb[ks], st);

    // ---- mask + online softmax. st element: key t = t0 + r + 8g, query = n.
    float p[8];
    float pm = -3.0e38f;
#pragma unroll
    for (int r = 0; r < 8; r++) {
      int t = t0 + r + 8 * g;
      float v = (st[r] * scale) * unscale;
      v = (t <= s0 + n) ? v : -3.0e38f;
      p[r] = v;
      pm = fmaxf(pm, v);
    }
    pm = fmaxf(pm, __shfl_xor(pm, 16, 32));
    float nm   = fmaxf(run_m, pm);
    float corr = __expf(run_m - nm);
    run_m = nm;
    float ls = 0.0f;
#pragma unroll
    for (int r = 0; r < 8; r++) { p[r] = __expf(p[r] - nm); ls += p[r]; }
    ls += __shfl_xor(ls, 16, 32);
    run_s = run_s * corr + ls;
#pragma unroll
    for (int dt = 0; dt < 4; dt++)
#pragma unroll
      for (int r = 0; r < 8; r++) acc[dt][r] *= corr;

    // ---- out^T += V^T x P  (4 key-slices of 4, 4 dim tiles of 16)
    const float* vbase = base + (size_t)t0 * QKVN + HD + DH;
#pragma unroll
    for (int sl = 0; sl < 4; sl++) {
      int j0 = sl * 4;
      // B-layout of P: b[v] = P[t = j0+v+2g][n]; source reg lives in the
      // other half-wave for one of the two groups -> shfl_xor(16).
      v2f pb;
      if (j0 < 8) {
        float o0 = __shfl_xor(p[j0 + 2], 16, 32);
        float o1 = __shfl_xor(p[j0 + 3], 16, 32);
        pb.x = (g == 0) ? p[j0]     : o0;
        pb.y = (g == 0) ? p[j0 + 1] : o1;
      } else {
        float o0 = __shfl_xor(p[j0 - 8],     16, 32);
        float o1 = __shfl_xor(p[j0 - 8 + 1], 16, 32);
        pb.x = (g == 0) ? o0 : p[j0 - 8 + 2];
        pb.y = (g == 0) ? o1 : p[j0 - 8 + 3];
      }
      // A = V^T: a[j] = V[t0+j0+2g+j][dt*16 + n]. Load all 4 operands,
      // then 4 independent WMMAs (distinct accumulators, back-to-back).
      const float* vp0 = vbase + (size_t)(j0 + 2 * g) * QKVN + n;
      v2f va[4];
#pragma unroll
      for (int dt = 0; dt < 4; dt++) {
        va[dt].x = vp0[dt * 16];
        va[dt].y = vp0[dt * 16 + QKVN];
      }
#pragma unroll
      for (int dt = 0; dt < 4; dt++) acc[dt] = wmma_f32(va[dt], pb, acc[dt]);
    }
  }

  float inv = 1.0f / run_s;
#pragma unroll
  for (int dt = 0; dt < 4; dt++)
#pragma unroll
    for (int r = 0; r < 8; r++) {
      int d = dt * 16 + r + 8 * g;   // out^T row = dim, col = query n
      out[((size_t)(b * SS + s0 + n)) * HD + head * DH + d] = acc[dt][r] * inv;
    }
}

// ---------------------------------------------------------------- launch
extern "C" void kernel_launch(void* const* d_in, const int* in_sizes, int n_in,
                              void* d_out, int out_size, void* d_ws, size_t ws_size,
                              hipStream_t stream) {
  (void)in_sizes; (void)n_in; (void)out_size; (void)ws_size;
  const int*   ids    = (const int*)d_in[0];
  const int*   pos    = (const int*)d_in[1];
  const float* wte    = (const float*)d_in[2];
  const float* wpe    = (const float*)d_in[3];
  const float* ln1w   = (const float*)d_in[4];
  const float* ln1b   = (const float*)d_in[5];
  const float* cattnw = (const float*)d_in[6];
  const float* cattnb = (const float*)d_in[7];
  const float* cprojw = (const float*)d_in[8];
  const float* cprojb = (const float*)d_in[9];
  const float* ln2w   = (const float*)d_in[10];
  const float* ln2b   = (const float*)d_in[11];
  const float* fcw    = (const float*)d_in[12];
  const float* fcb    = (const float*)d_in[13];
  const float* mprojw = (const float*)d_in[14];
  const float* mprojb = (const float*)d_in[15];
  const float* lnfw   = (const float*)d_in[16];
  const float* lnfb   = (const float*)d_in[17];

  float* ws  = (float*)d_ws;
  float* h   = ws;                          // TOK*HD
  float* x   = h + (size_t)TOK * HD;        // TOK*HD
  float* big = x + (size_t)TOK * HD;        // max(TOK*QKVN + TOK*HD, TOK*FF)
  float* qkvb = big;                        // TOK*QKVN
  float* ao   = big + (size_t)TOK * QKVN;   // TOK*HD (dead before ff is live)
  float* ff   = big;                        // TOK*FF

  embed_kernel<<<TOK, 256, 0, stream>>>(ids, pos, wte, wpe, h);

  for (int l = 0; l < LNUM; l++) {
    ln_kernel<<<TOK, 256, 0, stream>>>(h, ln1w + l * HD, ln1b + l * HD, x);
    gemm_kernel<<<dim3(QKVN / 64, TOK / 128), 256, 0, stream>>>(
        x, cattnw + (size_t)l * QKVN * HD, cattnb + (size_t)l * QKVN,
        nullptr, qkvb, QKVN, HD, 0);
    float unscale = (float)(l + 1);
    float scale   = 1.0f / (unscale * 8.0f);   // sqrt(D)=8
    attn_kernel<<<dim3(SS / 16, BB), 512, 0, stream>>>(qkvb, ao, scale, unscale);
    gemm_kernel<<<dim3(HD / 64, TOK / 128), 256, 0, stream>>>(
        ao, cprojw + (size_t)l * HD * HD, cprojb + (size_t)l * HD,
        h, h, HD, HD, 1);
    ln_kernel<<<TOK, 256, 0, stream>>>(h, ln2w + l * HD, ln2b + l * HD, x);
    gemm_kernel<<<dim3(FF / 64, TOK / 128), 256, 0, stream>>>(
        x, fcw + (size_t)l * FF * HD, fcb + (size_t)l * FF,
        nullptr, ff, FF, HD, 2);
    gemm_kernel<<<dim3(HD / 64, TOK / 128), 256, 0, stream>>>(
        ff, mprojw + (size_t)l * HD * FF, mprojb + (size_t)l * HD,
        h, h, HD, FF, 1);
  }

  ln_kernel<<<TOK, 256, 0, stream>>>(h, lnfw, lnfb, (float*)d_out);
}